// SpatialTemporalGather_Module_79534204387445
// MI455X (gfx1250) — compile-verified
//
#include <hip/hip_runtime.h>
#include <math.h>

typedef float v2f __attribute__((ext_vector_type(2)));
typedef float v8f __attribute__((ext_vector_type(8)));

#define B_TOT  8
#define C_DIM  512
#define K_DIM  19
#define N_SP   16384   // 128*128
#define CT     16      // channel tile (WMMA N)
#define WAVES  4       // waves per block
#define NSPLIT 4       // n-range splits across blocks
#define UNROLL 8       // WMMA k-steps per unrolled body (32 spatial elems)
#define PN_OFF 512     // float offset of normalized probs inside d_ws

// ---------------- Phase 0: zero output (harness poisons d_out) ----------------
__global__ void zero_out_kernel(float* __restrict__ out, int n) {
    int i = blockIdx.x * blockDim.x + threadIdx.x;
    if (i < n) out[i] = 0.0f;
}

// ---------------- Phase A: per-row softmax stats (max, 1/sum) ----------------
__global__ void softmax_stats_kernel(const float* __restrict__ probs,
                                     float* __restrict__ stats) {
    __shared__ float red[256];
    const int r = blockIdx.x;                       // row in [0, B_TOT*K_DIM)
    const float* p = probs + (size_t)r * N_SP;
    const int t = threadIdx.x;

    float m = -3.402823466e38f;
    for (int i = t; i < N_SP; i += 256) m = fmaxf(m, p[i]);
    red[t] = m; __syncthreads();
    for (int s = 128; s > 0; s >>= 1) {
        if (t < s) red[t] = fmaxf(red[t], red[t + s]);
        __syncthreads();
    }
    const float rowmax = red[0];
    __syncthreads();

    float sum = 0.0f;
    for (int i = t; i < N_SP; i += 256) sum += __expf(p[i] - rowmax);
    red[t] = sum; __syncthreads();
    for (int s = 128; s > 0; s >>= 1) {
        if (t < s) red[t] += red[t + s];
        __syncthreads();
    }
    if (t == 0) { stats[2 * r] = rowmax; stats[2 * r + 1] = 1.0f / red[0]; }
}

// ---------------- Phase B: materialize normalized probs into ws ----------------
__global__ void softmax_norm_kernel(const float* __restrict__ probs,
                                    const float* __restrict__ stats,
                                    float* __restrict__ pn, size_t total) {
    size_t i = (size_t)blockIdx.x * blockDim.x + threadIdx.x;
    if (i >= total) return;
    int r = (int)(i / N_SP);
    pn[i] = __expf(probs[i] - stats[2 * r]) * stats[2 * r + 1];
}

// ---------------- Phase C: fused pooling GEMM via V_WMMA_F32_16X16X4_F32 ------
// D[m=k][n=c] += sum_t A[k][t] * B[t][c], t = spatial chunk of 4.
// f32 16x4 A layout : lane<16 holds {t0,t1} of row m=lane ; lane>=16 holds {t2,t3}.
// f32 4x16 B layout : lane<16 holds {t0,t1} of col n=lane ; lane>=16 holds {t2,t3}.
// => both operands: one float2 load at row_base + n + (half ? 2 : 0).
__global__ void __launch_bounds__(WAVES * 32)
pooled_gemm_kernel(const float* __restrict__ feats,
                   const float* __restrict__ pn,
                   const int* __restrict__ clip_num,
                   float* __restrict__ out) {
    const int beta  = blockIdx.y;                   // flat batch 0..7
    const int ctile = blockIdx.x * CT;
    const int wave  = threadIdx.x >> 5;
    const int lane  = threadIdx.x & 31;
    const int half  = lane >> 4;
    const int lrow  = lane & 15;
    const int toff  = half * 2;

    const int   splits = clip_num[0] + 1;           // 4 for reference
    const int   bsz    = B_TOT / splits;            // 2
    const int   bb     = beta % bsz;                // batch index after split
    const float scale  = 1.0f / (float)splits;      // mean over splits

    constexpr int nper  = N_SP / (WAVES * NSPLIT);  // 1024 spatial elems / wave
    constexpr int NITER = nper / (4 * UNROLL);      // 32 outer iterations
    const int n0 = (blockIdx.z * WAVES + wave) * nper;

    // A rows (probs-k). Tile1 rows >= K_DIM alias row 0: garbage there only
    // affects D rows k>=19, which are never stored.
    const float* pa0 = pn + ((size_t)beta * K_DIM + lrow) * N_SP + toff + n0;
    const int    k1  = 16 + lrow;
    const float* pa1 = pn + ((size_t)beta * K_DIM + (k1 < K_DIM ? k1 : 0)) * N_SP
                          + toff + n0;
    // B rows (feats channels): streamed exactly once -> non-temporal
    const float* pb  = feats + ((size_t)beta * C_DIM + ctile + lrow) * N_SP
                             + toff + n0;

    v8f acc0 = {};
    v8f acc1 = {};

    // Uniform, compile-time trip count -> scalar loop, EXEC stays all-1s.
    for (int it = 0; it < NITER; ++it) {
        #pragma unroll
        for (int u = 0; u < UNROLL; ++u) {
            v2f a0 = *(const v2f*)(pa0 + 4 * u);
            v2f a1 = *(const v2f*)(pa1 + 4 * u);
            v2f bv = __builtin_nontemporal_load((const v2f*)(pb + 4 * u));
            acc0 = __builtin_amdgcn_wmma_f32_16x16x4_f32(
                       false, a0, false, bv, (short)0, acc0, false, false);
            acc1 = __builtin_amdgcn_wmma_f32_16x16x4_f32(
                       false, a1, false, bv, (short)0, acc1, false, false);
        }
        pa0 += 4 * UNROLL;
        pa1 += 4 * UNROLL;
        pb  += 4 * UNROLL;
    }

    // D layout: lane -> channel col = ctile + lrow ; VGPR v -> k = v + half*8.
    const int c = ctile + lrow;
    float* ob = out + ((size_t)bb * C_DIM + c) * K_DIM;
    #pragma unroll
    for (int v = 0; v < 8; ++v) {
        const int k = v + half * 8;                 // 0..15, always < K_DIM
        atomicAdd(ob + k, scale * acc0[v]);
        const int k2 = k + 16;                      // 16..31
        if (k2 < K_DIM) atomicAdd(ob + k2, scale * acc1[v]);
    }
}

extern "C" void kernel_launch(void* const* d_in, const int* in_sizes, int n_in,
                              void* d_out, int out_size, void* d_ws, size_t ws_size,
                              hipStream_t stream) {
    const float* feats = (const float*)d_in[0];
    const float* probs = (const float*)d_in[1];
    const int*   clip  = (const int*)d_in[2];
    float* out   = (float*)d_out;
    float* stats = (float*)d_ws;
    float* pnbuf = (float*)d_ws + PN_OFF;

    (void)in_sizes; (void)n_in; (void)ws_size;

    zero_out_kernel<<<(out_size + 255) / 256, 256, 0, stream>>>(out, out_size);

    softmax_stats_kernel<<<B_TOT * K_DIM, 256, 0, stream>>>(probs, stats);

    const size_t total = (size_t)B_TOT * K_DIM * N_SP;
    softmax_norm_kernel<<<(unsigned)((total + 255) / 256), 256, 0, stream>>>(
        probs, stats, pnbuf, total);

    dim3 grid(C_DIM / CT, B_TOT, NSPLIT);
    pooled_gemm_kernel<<<grid, WAVES * 32, 0, stream>>>(feats, pnbuf, clip, out);
}